// HiPPO_LegS_75548474736686
// MI455X (gfx1250) — compile-verified
//
#include <hip/hip_runtime.h>

// ---------------------------------------------------------------------------
// HiPPO-LegS scan on MI455X (gfx1250), f32 WMMA (V_WMMA_F32_16X16X4_F32).
//
// s_t = A_t s_{t-1} + u_t,  A_t: 128x128, s: 32x128, L=1024.
// Chunked linear-recurrence decomposition (chunk=16, 64 chunks):
//   1) compose:  P_c = A_{t0+15} ... A_{t0}           (parallel, WMMA 128^3)
//   2) localscan: q_c = chunk scan with zero init     (parallel)
//   3) chain:    h_{c+1} = P_c h_c + q_c  (1 block, 64 dependent steps)
//   4) replay:   rerun each chunk from h_c, write all outputs (parallel)
// Workspace: P (4MB) + q (1MB) + h (1MB) = 6MB, fully overwritten every call.
//
//  * compose keeps the running product TRANSPOSED + SKEWED in LDS
//    (MT[j][ (i+4j)&127 ] = M[i][j]) -> B fragments become single
//    bank-conflict-free ds_load_b64, write-back becomes 2x b128 stores.
//  * software-pipelined global fragment loads in compose & scan.
//  * gfx1250 async LDS<->global ops (ASYNCcnt) used for input staging and
//    for streaming per-step outputs.  Round-3 fix: the async builtins take
//    (as1 v4i*, as3 v4i*, imm, imm) — cast pointee type accordingly.
// ---------------------------------------------------------------------------

typedef float v2f __attribute__((ext_vector_type(2)));
typedef float v4f __attribute__((ext_vector_type(4)));
typedef float v8f __attribute__((ext_vector_type(8)));
typedef int   vi4 __attribute__((vector_size(16)));

#define NSTATE 128
#define NBATCH 32
#define LSEQ   1024
#define CHUNK  16
#define NCHUNK 64
#define SPAD   132  // padded LDS row stride for the scan state (conflict-free)

// skewed column index for the 128x128 transposed product held in LDS
#define SK(col, row) ((((col) + 4 * (row))) & 127)

#if __has_builtin(__builtin_amdgcn_global_load_async_to_lds_b128) && \
    __has_builtin(__builtin_amdgcn_global_store_async_from_lds_b128)
#define ASYNC_OK 1
#endif

static __device__ __forceinline__ void async_load_b128(const float* gsrc,
                                                       float* ldst) {
#ifdef ASYNC_OK
  __builtin_amdgcn_global_load_async_to_lds_b128(
      (__attribute__((address_space(1))) vi4*)gsrc,
      (__attribute__((address_space(3))) vi4*)ldst, 0, 0);
#else
  *(v4f*)ldst = *(const v4f*)gsrc;
#endif
}

static __device__ __forceinline__ void async_store_b128(float* gdst,
                                                        const float* lsrc) {
#ifdef ASYNC_OK
  __builtin_amdgcn_global_store_async_from_lds_b128(
      (__attribute__((address_space(1))) vi4*)gdst,
      (__attribute__((address_space(3))) vi4*)lsrc, 0, 0);
#else
  *(v4f*)gdst = *(const v4f*)lsrc;
#endif
}

static __device__ __forceinline__ void wait_async0() {
#ifdef ASYNC_OK
#if __has_builtin(__builtin_amdgcn_s_wait_asynccnt)
  __builtin_amdgcn_s_wait_asynccnt(0);
#else
  asm volatile("s_wait_asynccnt 0" ::: "memory");
#endif
#endif
}

// D = A(16x4) * B(4x16) + C, f32. Layouts (wave32, g = lane>>4, r = lane&15):
//   A frag a[v] = A[m=r][k = k0 + 2g + v]   (contiguous in k -> b64 load)
//   B frag b[v] = B[k = k0 + 2g + v][n=r]
//   D     d[v] = D[m = v + 8g][n = r]
static __device__ __forceinline__ v8f wmma_f32(v2f a, v2f b, v8f c) {
  return __builtin_amdgcn_wmma_f32_16x16x4_f32(false, a, false, b, (short)0, c,
                                               false, false);
}

// ---------------------------------------------------------------------------
// Pass 1: P_c = A_{t0+15} * ... * A_{t0}.  One block per chunk, 512 threads.
// Product MT (transposed + skewed, 64KB) in LDS; A_t fragments stream from
// global (L2-resident).  16 waves, each owns a 2x2 block of 16x16 tiles.
// ---------------------------------------------------------------------------
__global__ __launch_bounds__(512) void hippo_compose_kernel(
    const float* __restrict__ A, float* __restrict__ P) {
  __shared__ float MT[NSTATE * NSTATE];  // MT[j*128 + SK(i,j)] = M[i][j]
  const int tid  = threadIdx.x;
  const int lane = tid & 31;
  const int w    = tid >> 5;
  const int g    = lane >> 4;
  const int r    = lane & 15;
  const int c    = blockIdx.x;
  const long t0  = (long)c * CHUNK;

  // MT = A[t0] (transposed + skewed)
  const float* A0 = A + t0 * (NSTATE * NSTATE);
  for (int e = tid; e < NSTATE * NSTATE; e += 512) {
    const int i = e >> 7, j = e & 127;
    MT[j * NSTATE + SK(i, j)] = A0[e];
  }
  __syncthreads();

  const int tm2 = w >> 2;                      // 0..3
  const int tn2 = w & 3;                       // 0..3
  const int m0  = 16 * (2 * tm2 + 0) + r;      // A-frag rows (global)
  const int m1  = 16 * (2 * tm2 + 1) + r;
  const int n0  = 16 * (2 * tn2 + 0) + r;      // B-frag columns (LDS rows)
  const int n1  = 16 * (2 * tn2 + 1) + r;

  for (int st = 1; st < CHUNK; ++st) {
    const float* At = A + (t0 + st) * (NSTATE * NSTATE);
    {  // prefetch this wave's rows of the *next* A_t
      long tnx = t0 + st + 1;
      if (tnx > LSEQ - 1) tnx = LSEQ - 1;
      const float* Ap = A + tnx * (NSTATE * NSTATE);
      __builtin_prefetch(Ap + m0 * NSTATE, 0, 3);
      __builtin_prefetch(Ap + m1 * NSTATE, 0, 3);
    }
    v8f acc00 = {}, acc01 = {}, acc10 = {}, acc11 = {};
    // software-pipelined global A-fragment loads
    v2f a0 = *(const v2f*)(At + m0 * NSTATE + 2 * g);
    v2f a1 = *(const v2f*)(At + m1 * NSTATE + 2 * g);
    for (int k0 = 0; k0 < NSTATE; k0 += 4) {
      const int kk = k0 + 2 * g;
      const int kn = ((k0 + 4) & 127) + 2 * g;  // wraps on last iter (benign)
      v2f na0 = *(const v2f*)(At + m0 * NSTATE + kn);
      v2f na1 = *(const v2f*)(At + m1 * NSTATE + kn);
      // B fragments: single conflict-free b64 from skewed MT
      v2f b0 = *(const v2f*)(&MT[n0 * NSTATE + SK(kk, n0)]);
      v2f b1 = *(const v2f*)(&MT[n1 * NSTATE + SK(kk, n1)]);
      acc00 = wmma_f32(a0, b0, acc00);
      acc01 = wmma_f32(a0, b1, acc01);
      acc10 = wmma_f32(a1, b0, acc10);
      acc11 = wmma_f32(a1, b1, acc11);
      a0 = na0;
      a1 = na1;
    }
    __syncthreads();  // all reads of M_old done
    // write back: D[m][n] -> MT[n][SK(m,n)]; m = tile_base + v + 8g is
    // contiguous in v -> two aligned b128 stores per tile.
    {
      const int mb0 = 16 * (2 * tm2 + 0) + 8 * g;
      const int mb1 = 16 * (2 * tm2 + 1) + 8 * g;
      v4f t;
      int cbase;
#define WB_TILE(ACC, MB, NN)                                   \
  cbase = ((MB) + 4 * (NN)) & 127;                             \
  t = (v4f){ACC[0], ACC[1], ACC[2], ACC[3]};                   \
  *(v4f*)(&MT[(NN)*NSTATE + cbase]) = t;                       \
  t = (v4f){ACC[4], ACC[5], ACC[6], ACC[7]};                   \
  *(v4f*)(&MT[(NN)*NSTATE + ((cbase + 4) & 127)]) = t;
      WB_TILE(acc00, mb0, n0)
      WB_TILE(acc01, mb0, n1)
      WB_TILE(acc10, mb1, n0)
      WB_TILE(acc11, mb1, n1)
#undef WB_TILE
    }
    __syncthreads();
  }

  // store P_c row-major: P_c[i][j] = MT[j][SK(i,j)]
  float* Pc = P + (long)c * (NSTATE * NSTATE);
  for (int e = tid; e < NSTATE * NSTATE; e += 512) {
    const int i = e >> 7, j = e & 127;
    Pc[e] = MT[j * NSTATE + SK(i, j)];
  }
}

// ---------------------------------------------------------------------------
// Passes 2 & 4: chunk-local scan.  s_new[b,i] = sum_j A_t[i,j] s[b,j] + u[b,i]
// with u[t,b,n] = inputs[t,b] * Bst[t,n].  16 waves, one 16x16 tile each.
// B operand = A_t^T -> contiguous b64 global loads along k (L2-resident).
// Outputs are streamed LDS->global with async stores (overlap next GEMM).
// ---------------------------------------------------------------------------
__global__ __launch_bounds__(512) void hippo_scan_kernel(
    const float* __restrict__ A, const float* __restrict__ inp,
    const float* __restrict__ Bst, const float* __restrict__ init,
    float* __restrict__ out, float* __restrict__ qout, int has_init,
    int write_out) {
  __shared__ float s[NBATCH * SPAD];     // 16.9KB state, padded rows
  __shared__ float uin[CHUNK * NBATCH];  // inputs[t0..t0+15][0..31]
  __shared__ float bin[CHUNK * NSTATE];  // Bst[t0..t0+15][0..127]
  const int tid  = threadIdx.x;
  const int lane = tid & 31;
  const int w    = tid >> 5;
  const int g    = lane >> 4;
  const int r    = lane & 15;
  const int c    = blockIdx.x;
  const long t0  = (long)c * CHUNK;

  // --- stage inputs (async where available) ---
  if (tid < CHUNK * NBATCH / 4)  // 128 x b128
    async_load_b128(inp + t0 * NBATCH + tid * 4, uin + tid * 4);
  // 512 x b128 for Bst chunk
  async_load_b128(Bst + t0 * NSTATE + tid * 4, bin + tid * 4);
  if (has_init) {
    // 1024 x b128, 2 per thread, into padded-state layout
#pragma unroll
    for (int q4 = 0; q4 < 2; ++q4) {
      const int g4 = tid + q4 * 512;
      const int b = g4 >> 5, n4 = g4 & 31;
      async_load_b128(init + (long)c * NBATCH * NSTATE + b * NSTATE + n4 * 4,
                      s + b * SPAD + n4 * 4);
    }
  } else {
    for (int i = tid; i < NBATCH * NSTATE; i += 512) {
      const int b = i >> 7, n = i & 127;
      s[b * SPAD + n] = 0.0f;
    }
  }
  wait_async0();
  __syncthreads();

  const int tm = w >> 3;  // 0..1  (batch tile)
  const int tn = w & 7;   // 0..7  (state tile)
  // this thread's two b128 output groups (for async LDS->global streaming)
  const int ob0 = tid >> 5, on0 = (tid & 31) * 4;            // group 0
  const int ob1 = (tid + 512) >> 5, on1 = (tid & 31) * 4;    // group 1

  for (int st = 0; st < CHUNK; ++st) {
    const float* At = A + (t0 + st) * (NSTATE * NSTATE);
    {
      long tnx = t0 + st + 1;
      if (tnx > LSEQ - 1) tnx = LSEQ - 1;
      __builtin_prefetch(A + tnx * (NSTATE * NSTATE) + (16 * tn + r) * NSTATE,
                         0, 3);
    }
    v8f acc = {};
    v2f b = *(const v2f*)(At + (16 * tn + r) * NSTATE + 2 * g);
    for (int k0 = 0; k0 < NSTATE; k0 += 4) {
      const int kk = k0 + 2 * g;
      const int kn = ((k0 + 4) & 127) + 2 * g;
      v2f nb = *(const v2f*)(At + (16 * tn + r) * NSTATE + kn);
      v2f a = *(const v2f*)(&s[(16 * tm + r) * SPAD + kk]);  // s[b][k]
      acc = wmma_f32(a, b, acc);
      b = nb;
    }
    // + u_t
    const float bv = bin[st * NSTATE + 16 * tn + r];
#pragma unroll
    for (int v = 0; v < 8; ++v)
      acc[v] += uin[st * NBATCH + 16 * tm + v + 8 * g] * bv;
    __syncthreads();   // everyone done reading s_old
    wait_async0();     // our previous step's async output stores done
#pragma unroll
    for (int v = 0; v < 8; ++v)
      s[(16 * tm + v + 8 * g) * SPAD + 16 * tn + r] = acc[v];
    __syncthreads();
    if (write_out) {   // stream s -> out[t] asynchronously from LDS
      float* ot = out + (t0 + st) * (NBATCH * NSTATE);
      async_store_b128(ot + ob0 * NSTATE + on0, s + ob0 * SPAD + on0);
      async_store_b128(ot + ob1 * NSTATE + on1, s + ob1 * SPAD + on1);
    }
  }

  if (qout != nullptr) {
    for (int i = tid; i < NBATCH * NSTATE; i += 512) {
      const int b = i >> 7, n = i & 127;
      qout[(long)c * NBATCH * NSTATE + i] = s[b * SPAD + n];
    }
  }
  // kernel end waits out all async counters implicitly (S_ENDPGM)
}

// ---------------------------------------------------------------------------
// Pass 3: h_{c+1} = P_c h_c + q_c, storing every chunk-start state h_c.
// Single block, 64 dependent iterations (cheap: 32x128x128 GEMM each).
// ---------------------------------------------------------------------------
__global__ __launch_bounds__(512) void hippo_chain_kernel(
    const float* __restrict__ P, const float* __restrict__ q,
    float* __restrict__ h) {
  __shared__ float s[NBATCH * SPAD];
  const int tid  = threadIdx.x;
  const int lane = tid & 31;
  const int w    = tid >> 5;
  const int g    = lane >> 4;
  const int r    = lane & 15;
  const int tm   = w >> 3;
  const int tn   = w & 7;

  for (int i = tid; i < NBATCH * NSTATE; i += 512) {
    const int b = i >> 7, n = i & 127;
    s[b * SPAD + n] = 0.0f;  // s_0 = 0
  }
  __syncthreads();

  for (int c = 0; c < NCHUNK; ++c) {
    // record chunk-start state
    for (int i = tid; i < NBATCH * NSTATE; i += 512) {
      const int b = i >> 7, n = i & 127;
      h[(long)c * NBATCH * NSTATE + i] = s[b * SPAD + n];
    }
    const float* Pc = P + (long)c * (NSTATE * NSTATE);
    const float* qc = q + (long)c * (NBATCH * NSTATE);
    v8f acc = {};
    v2f b = *(const v2f*)(Pc + (16 * tn + r) * NSTATE + 2 * g);
    for (int k0 = 0; k0 < NSTATE; k0 += 4) {
      const int kk = k0 + 2 * g;
      const int kn = ((k0 + 4) & 127) + 2 * g;
      v2f nb = *(const v2f*)(Pc + (16 * tn + r) * NSTATE + kn);
      v2f a = *(const v2f*)(&s[(16 * tm + r) * SPAD + kk]);
      acc = wmma_f32(a, b, acc);
      b = nb;
    }
#pragma unroll
    for (int v = 0; v < 8; ++v)
      acc[v] += qc[(16 * tm + v + 8 * g) * NSTATE + 16 * tn + r];
    __syncthreads();
#pragma unroll
    for (int v = 0; v < 8; ++v)
      s[(16 * tm + v + 8 * g) * SPAD + 16 * tn + r] = acc[v];
    __syncthreads();
  }
}

extern "C" void kernel_launch(void* const* d_in, const int* in_sizes, int n_in,
                              void* d_out, int out_size, void* d_ws,
                              size_t ws_size, hipStream_t stream) {
  const float* inp = (const float*)d_in[0];  // (1024, 32)
  const float* A   = (const float*)d_in[1];  // (1024, 128, 128)
  const float* Bst = (const float*)d_in[2];  // (1024, 128)
  float* out = (float*)d_out;                // (1024, 32, 128)

  float* P = (float*)d_ws;                          // 4MB
  float* q = P + (size_t)NCHUNK * NSTATE * NSTATE;  // 1MB
  float* h = q + (size_t)NCHUNK * NBATCH * NSTATE;  // 1MB

  hippo_compose_kernel<<<NCHUNK, 512, 0, stream>>>(A, P);
  hippo_scan_kernel<<<NCHUNK, 512, 0, stream>>>(A, inp, Bst, nullptr, nullptr,
                                                q, 0, 0);
  hippo_chain_kernel<<<1, 512, 0, stream>>>(P, q, h);
  hippo_scan_kernel<<<NCHUNK, 512, 0, stream>>>(A, inp, Bst, h, out, nullptr,
                                                1, 1);
}